// DSTFT_70239895158907
// MI455X (gfx1250) — compile-verified
//
#include <hip/hip_runtime.h>
#include <math.h>

#define N_       256
#define T_       16383
#define B_       8
#define L_       2097152LL
#define F_       129
#define NFT      9              // freq tiles: f = 0..143 (covers F_=129; rest by symmetry)
#define GRP_PB   512            // groups of 32 frames per batch (512*32 = 16384 >= T_)
#define PI_F     3.14159265358979323846f

typedef __attribute__((ext_vector_type(2))) float v2f;
typedef __attribute__((ext_vector_type(8))) float v8f;

// ---- output layout (floats), concatenated in reference return order ----
// spec  : [0, SPEC)                       B*F*T
// stft  : [SPEC, SPEC+2*BNT)              complex64 interleaved (re,im)
// real  : [SPEC+2*BNT, SPEC+3*BNT)
// imag  : [SPEC+3*BNT, SPEC+4*BNT)
// phase : [SPEC+4*BNT, SPEC+4*BNT+SPEC)
static const long long SPEC_SZ = (long long)B_ * F_ * T_;       // 16,907,256
static const long long BNT     = (long long)B_ * N_ * T_;       // 33,552,384

// DFT coefficient table pre-swizzled into WMMA B-fragment layout.
// Entry [ft][ks][lane] is float4: (re_j0, re_j1, im_j0, im_j1) where
// K = ks*4 + j + (lane>=16 ? 2 : 0), Nf = ft*16 + (lane&15),
// coeff = exp(-2*pi*i * K * Nf / 256).  (K*Nf) mod 256 is an exact reduction.
__global__ void coeff_init_kernel(float4* __restrict__ tbl) {
    int idx = blockIdx.x * blockDim.x + threadIdx.x;
    const int total = NFT * 64 * 32;
    if (idx >= total) return;
    int lane = idx & 31;
    int ks   = (idx >> 5) & 63;
    int ft   = idx / (64 * 32);
    int nf   = ft * 16 + (lane & 15);
    int kb   = ks * 4 + ((lane >> 4) << 1);
    int m0 = (kb * nf) & 255;
    int m1 = ((kb + 1) * nf) & 255;
    float a0 = -2.0f * PI_F * (float)m0 * (1.0f / 256.0f);
    float a1 = -2.0f * PI_F * (float)m1 * (1.0f / 256.0f);
    float4 v;
    v.x = cosf(a0); v.y = cosf(a1);   // real part of exp(-i*theta)
    v.z = sinf(a0); v.w = sinf(a1);   // imag part of exp(-i*theta)
    tbl[idx] = v;
}

#define A_STRIDE     260                     // padded row stride (floats) -> conflict-free ds_load_b64
#define A_SIZE       (32 * A_STRIDE)         // 32 frames per wave
#define TILE_STRIDE  17                      // padded 16x16 tile row stride
#define TILE_SIZE    (16 * TILE_STRIDE)      // 272 floats

__launch_bounds__(32, 1)
__global__ void dstft_kernel(const float* __restrict__ x,
                             const float* __restrict__ win_length,
                             const float* __restrict__ strides,
                             const float* __restrict__ win_pow,
                             const float4* __restrict__ coeff,
                             float* __restrict__ out)
{
    __shared__ float lds[A_SIZE + 2 * TILE_SIZE];   // ~35.5 KB
    const int lane = threadIdx.x;
    const int grp  = blockIdx.x;                    // 0 .. B_*GRP_PB-1
    const int b    = grp / GRP_PB;
    const int gg   = grp - b * GRP_PB;
    const int t0   = gg * 32;                       // first frame of this 32-frame group

    float* A   = lds;
    float* tRe = A + A_SIZE;
    float* tIm = tRe + TILE_SIZE;

    // learned scalar params (clamp transforms)
    const float wl = fminf(fmaxf(win_length[0], (float)N_ / 20.0f), (float)N_);
    const float st = fminf(fmaxf(strides[0], 0.0f), 256.0f);
    const float wp = win_pow[0];
    const float hi = ceilf((255.0f + wl) * 0.5f);
    const float lo = floorf((255.0f - wl) * 0.5f);
    const float csh = (wl - 255.0f) * 0.5f;         // (wl - N + 1)/2
    const float wfrq = 2.0f * PI_F / wl;

    // ---------- phase 1: learnable Hann taper * gathered frame -> LDS A rows ----------
    // lane l owns frame t0+l entirely (256 taps) -> no cross-lane reduction needed
    {
        const int  t      = t0 + lane;
        const bool tvalid = (t < T_);
        const float fpos  = (t == 0) ? 0.0f : (float)t * st;  // == cumsum of equal strides (exact)
        const float ffl   = floorf(fpos);
        const float frac  = fpos - ffl;
        const long long i0 = (long long)ffl;
        float* Arow = A + lane * A_STRIDE;

        float sum = 0.0f;
        #pragma unroll 4
        for (int n = 0; n < 256; ++n) {
            float base = (float)n - frac;
            float tap = 0.0f;
            if (base < hi && base > lo)             // zero outside support (matches reference)
                tap = 0.5f - 0.5f * cosf(wfrq * (base + csh));
            Arow[n] = tap;
            sum += tap;
        }
        const float rnorm = 1.0f / sum;
        const float* xb = x + (long long)b * L_;
        #pragma unroll 4
        for (int n = 0; n < 256; ++n) {
            float tap = Arow[n] * rnorm;
            if (wp != 1.0f) tap = powf(tap, wp);
            long long gi = i0 + (long long)n;
            float xv = (tvalid && gi >= 0 && gi < L_) ? xb[gi] : 0.0f;
            Arow[n] = tap * xv;
        }
    }
    // single wave, wave-private LDS; DS ops are in-order within a wave -> no barrier

    // per-lane epilogue frame mapping (used after LDS transpose): t = t0 + s*16 + (lane&15)
    const int  half = lane >> 4;
    const int  tl   = lane & 15;
    float fr2[2], phiS[2], phiC[2];
    bool  tv[2];
    int   tcur[2];
    #pragma unroll
    for (int s = 0; s < 2; ++s) {
        tcur[s] = t0 + s * 16 + tl;
        tv[s]   = (tcur[s] < T_);
        float fp = (tcur[s] == 0) ? 0.0f : (float)tcur[s] * st;
        fr2[s] = fp - floorf(fp);
        __sincosf(2.0f * PI_F * fr2[s], &phiS[s], &phiC[s]);   // arg < 2*pi -> native is accurate
    }

    const int aoff0 = tl * A_STRIDE;                 // subtile 0 A-fragment row
    const int aoff1 = (16 + tl) * A_STRIDE;          // subtile 1 A-fragment row
    const int koff  = half << 1;                     // lanes >= 16 take K+2,K+3

    // ---------- phase 2: fp32 WMMA GEMM over 9 frequency tiles, 2 frame subtiles ----------
    for (int ft = 0; ft < NFT; ++ft) {
        v8f accRe0 = {0.f,0.f,0.f,0.f,0.f,0.f,0.f,0.f};
        v8f accIm0 = {0.f,0.f,0.f,0.f,0.f,0.f,0.f,0.f};
        v8f accRe1 = {0.f,0.f,0.f,0.f,0.f,0.f,0.f,0.f};
        v8f accIm1 = {0.f,0.f,0.f,0.f,0.f,0.f,0.f,0.f};
        const float4* cp = coeff + (long long)(ft * 64) * 32 + lane;

        #pragma unroll 8
        for (int ks = 0; ks < 64; ++ks) {
            float4 c4 = cp[ks * 32];                           // shared B fragments (re,im)
            v2f a0 = *(const v2f*)(A + aoff0 + ks * 4 + koff); // A fragment, subtile 0
            v2f a1 = *(const v2f*)(A + aoff1 + ks * 4 + koff); // A fragment, subtile 1
            v2f bre; bre.x = c4.x; bre.y = c4.y;
            v2f bim; bim.x = c4.z; bim.y = c4.w;
            accRe0 = __builtin_amdgcn_wmma_f32_16x16x4_f32(
                false, a0, false, bre, (short)0, accRe0, false, false);
            accIm0 = __builtin_amdgcn_wmma_f32_16x16x4_f32(
                false, a0, false, bim, (short)0, accIm0, false, false);
            accRe1 = __builtin_amdgcn_wmma_f32_16x16x4_f32(
                false, a1, false, bre, (short)0, accRe1, false, false);
            accIm1 = __builtin_amdgcn_wmma_f32_16x16x4_f32(
                false, a1, false, bim, (short)0, accIm1, false, false);
        }

        // per subtile: transpose through padded LDS, then coalesced t-contiguous stores
        #pragma unroll
        for (int s = 0; s < 2; ++s) {
            const v8f aR = s ? accRe1 : accRe0;
            const v8f aI = s ? accIm1 : accIm0;
            #pragma unroll
            for (int v = 0; v < 8; ++v) {
                int M = v + 8 * half;
                tRe[M * TILE_STRIDE + tl] = aR[v];
                tIm[M * TILE_STRIDE + tl] = aI[v];
            }
            const float fr = fr2[s], cphi = phiC[s], sphi = phiS[s];
            const bool  tok = tv[s];
            const long long tt64 = (long long)tcur[s];
            #pragma unroll
            for (int it = 0; it < 8; ++it) {
                int fl = half + 2 * it;              // freq row within tile
                int f  = ft * 16 + fl;
                float re0 = tRe[tl * TILE_STRIDE + fl];
                float im0 = tIm[tl * TILE_STRIDE + fl];
                // sub-sample phase shift: exp(+2*pi*i * frac * f / N); arg < 2*pi
                float th = (2.0f * PI_F / 256.0f) * fr * (float)f;
                float sn, cs;
                __sincosf(th, &sn, &cs);
                float re = re0 * cs - im0 * sn;
                float im = re0 * sn + im0 * cs;
                if (tok) {
                    long long sidx = ((long long)(b * N_ + f)) * T_ + tt64;
                    if (f < F_) {
                        long long pidx = ((long long)(b * F_ + f)) * T_ + tt64;
                        out[pidx] = sqrtf(re * re + im * im) + 1.1920928955078125e-07f;
                        out[SPEC_SZ + 4 * BNT + pidx] = atan2f(im, re);
                    }
                    out[SPEC_SZ + 2 * sidx]     = re;    // stft (interleaved complex)
                    out[SPEC_SZ + 2 * sidx + 1] = im;
                    out[SPEC_SZ + 2 * BNT + sidx] = re;  // real
                    out[SPEC_SZ + 3 * BNT + sidx] = im;  // imag
                    // mirror: stft[256-f] = conj(stft[f]) * exp(2*pi*i*frac)
                    if (f >= 1 && f <= 112) {
                        int f2 = N_ - f;                 // 144..255
                        float re2 = re * cphi + im * sphi;
                        float im2 = re * sphi - im * cphi;
                        long long s2 = ((long long)(b * N_ + f2)) * T_ + tt64;
                        out[SPEC_SZ + 2 * s2]     = re2;
                        out[SPEC_SZ + 2 * s2 + 1] = im2;
                        out[SPEC_SZ + 2 * BNT + s2] = re2;
                        out[SPEC_SZ + 3 * BNT + s2] = im2;
                    }
                }
            }
        }
    }
}

extern "C" void kernel_launch(void* const* d_in, const int* in_sizes, int n_in,
                              void* d_out, int out_size, void* d_ws, size_t ws_size,
                              hipStream_t stream) {
    (void)in_sizes; (void)n_in; (void)out_size; (void)ws_size;
    const float* x  = (const float*)d_in[0];
    const float* wl = (const float*)d_in[1];
    const float* st = (const float*)d_in[2];
    const float* wp = (const float*)d_in[3];
    float4* coeff = (float4*)d_ws;                 // needs NFT*64*32*16 B = 294,912 B

    const int total = NFT * 64 * 32;
    coeff_init_kernel<<<(total + 255) / 256, 256, 0, stream>>>(coeff);

    const int groups = B_ * GRP_PB;                // 4096 blocks, 32 threads each
    dstft_kernel<<<groups, 32, 0, stream>>>(
        x, wl, st, wp, (const float4*)coeff, (float*)d_out);
}